// HeteroGNNModel_43576738185766
// MI455X (gfx1250) — compile-verified
//
#include <hip/hip_runtime.h>

// ---------------------------------------------------------------------------
// HeteroGNN forward on MI455X (gfx1250, wave32).
// All GEMMs use V_WMMA_F32_16X16X4_F32 (native f32 matrix op) so numerics
// match the f32 reference. Problem is L2-resident (~103MB workspace < 192MB
// L2); the edge scatter is the bottleneck, GEMMs ride the matrix pipes.
// ---------------------------------------------------------------------------

typedef __attribute__((ext_vector_type(2))) float v2f;
typedef __attribute__((ext_vector_type(8))) float v8f;

#define N_LOC 100000
#define N_EVT 200000
#define NEDGE 1000000
#define F_IN  128
#define HDIM  64
#define HHALF 32

__device__ __forceinline__ v8f wmma_f32(v2f a, v2f b, v8f c) {
  // (neg_a, A, neg_b, B, c_mod, C, reuse_a, reuse_b)
  return __builtin_amdgcn_wmma_f32_16x16x4_f32(false, a, false, b, (short)0, c,
                                               false, false);
}

// ---------------------------------------------------------------------------
// Y[N, 64] = relu(X[N, K] @ W[K, 64] + b).  Block = 256 thr = 8 waves.
// Wave (rowTile = w>>2 in 0..1, colTile = w&3 in 0..3) -> block covers 32 rows.
// N must be a multiple of 32 (100000 and 200000 both are).
// ---------------------------------------------------------------------------
__global__ __launch_bounds__(256) void proj_gemm_relu(
    const float* __restrict__ X, const float* __restrict__ W,
    const float* __restrict__ bias, float* __restrict__ Y, int K) {
  const int lane = threadIdx.x & 31;
  const int wave = threadIdx.x >> 5;
  const int rowBase = blockIdx.x * 32 + (wave >> 2) * 16;
  const int colBase = (wave & 3) * 16;
  const int m = rowBase + (lane & 15);
  const int n = colBase + (lane & 15);
  const int koff = (lane >> 4) << 1;  // 0 or 2

  const float* xrow = X + (size_t)m * K;
  v8f c = {};
#pragma unroll 8
  for (int k = 0; k < K; k += 4) {
    v2f a = *(const v2f*)(xrow + k + koff);  // contiguous K pair -> b64 load
    v2f b;
    b.x = W[(size_t)(k + koff) * HDIM + n];
    b.y = W[(size_t)(k + koff + 1) * HDIM + n];
    c = wmma_f32(a, b, c);
  }
  const float bn = bias[n];
  const int half = lane >> 4;
#pragma unroll
  for (int v = 0; v < 8; ++v) {
    const int row = rowBase + v + 8 * half;
    const float val = c[v] + bn;
    Y[(size_t)row * HDIM + n] = val > 0.f ? val : 0.f;
  }
}

// ---------------------------------------------------------------------------
// Zero msg_sum + cnt (contiguous region).
// ---------------------------------------------------------------------------
__global__ void zero_kernel(float* __restrict__ p, size_t n) {
  const size_t i = (size_t)blockIdx.x * blockDim.x + threadIdx.x;
  if (i < n) p[i] = 0.f;
}

// ---------------------------------------------------------------------------
// Edge scatter: 64 threads per edge, one feature each; f32 atomics into msg.
// ---------------------------------------------------------------------------
__global__ __launch_bounds__(256) void scatter_edges(
    const long long* __restrict__ ei, const float* __restrict__ evt_h,
    float* __restrict__ msg, float* __restrict__ cnt) {
  const long long t = (long long)blockIdx.x * blockDim.x + threadIdx.x;
  const int e = (int)(t >> 6);
  const int f = (int)(t & 63);
  if (e >= NEDGE) return;
  const int src = (int)ei[e];
  const int dst = (int)ei[NEDGE + e];
  atomicAdd(&msg[(size_t)dst * HDIM + f], evt_h[(size_t)src * HDIM + f]);
  if (f == 0) atomicAdd(&cnt[dst], 1.0f);
}

// ---------------------------------------------------------------------------
// loc_h2 = relu((msg/max(cnt,1)) @ W_l + b_l + loc_h @ W_r), written IN PLACE
// over the msg buffer (barrier before store; each block owns its 32 rows).
// ---------------------------------------------------------------------------
__global__ __launch_bounds__(256) void sage_combine(
    const float* msg, const float* __restrict__ cnt, const float* __restrict__ loc_h,
    const float* __restrict__ W_l, const float* __restrict__ b_l,
    const float* __restrict__ W_r, float* Y) {
  const int lane = threadIdx.x & 31;
  const int wave = threadIdx.x >> 5;
  const int rowBase = blockIdx.x * 32 + (wave >> 2) * 16;
  const int colBase = (wave & 3) * 16;
  const int m = rowBase + (lane & 15);
  const int n = colBase + (lane & 15);
  const int koff = (lane >> 4) << 1;

  const float inv = 1.0f / fmaxf(cnt[m], 1.0f);
  v8f c = {};
#pragma unroll 8
  for (int k = 0; k < HDIM; k += 4) {  // agg @ W_l (scale A rows by 1/cnt)
    v2f a = *(const v2f*)(msg + (size_t)m * HDIM + k + koff);
    a.x *= inv; a.y *= inv;
    v2f b;
    b.x = W_l[(size_t)(k + koff) * HDIM + n];
    b.y = W_l[(size_t)(k + koff + 1) * HDIM + n];
    c = wmma_f32(a, b, c);
  }
#pragma unroll 8
  for (int k = 0; k < HDIM; k += 4) {  // + loc_h @ W_r (same accumulator)
    v2f a = *(const v2f*)(loc_h + (size_t)m * HDIM + k + koff);
    v2f b;
    b.x = W_r[(size_t)(k + koff) * HDIM + n];
    b.y = W_r[(size_t)(k + koff + 1) * HDIM + n];
    c = wmma_f32(a, b, c);
  }
  __syncthreads();  // all waves done reading block-owned rows before in-place store
  const float bn = b_l[n];
  const int half = lane >> 4;
#pragma unroll
  for (int v = 0; v < 8; ++v) {
    const int row = rowBase + v + 8 * half;
    const float val = c[v] + bn;
    Y[(size_t)row * HDIM + n] = val > 0.f ? val : 0.f;
  }
}

// ---------------------------------------------------------------------------
// h1[N_LOC, 32] = relu(loc_h2 @ W_h1 + b_h1).  8 waves = 4 rowTiles x 2
// colTiles -> 64 rows/block; clamp/guard the 100000 % 64 tail.
// ---------------------------------------------------------------------------
__global__ __launch_bounds__(256) void head1_gemm_relu(
    const float* __restrict__ X, const float* __restrict__ W,
    const float* __restrict__ bias, float* __restrict__ Y) {
  const int lane = threadIdx.x & 31;
  const int wave = threadIdx.x >> 5;
  const int rowBase = blockIdx.x * 64 + (wave >> 1) * 16;
  const int colBase = (wave & 1) * 16;
  const int m = rowBase + (lane & 15);
  const int mc = m < N_LOC ? m : (N_LOC - 1);  // clamp for safe loads
  const int n = colBase + (lane & 15);
  const int koff = (lane >> 4) << 1;

  v8f c = {};
#pragma unroll 8
  for (int k = 0; k < HDIM; k += 4) {
    v2f a = *(const v2f*)(X + (size_t)mc * HDIM + k + koff);
    v2f b;
    b.x = W[(size_t)(k + koff) * HHALF + n];
    b.y = W[(size_t)(k + koff + 1) * HHALF + n];
    c = wmma_f32(a, b, c);
  }
  const float bn = bias[n];
  const int half = lane >> 4;
#pragma unroll
  for (int v = 0; v < 8; ++v) {
    const int row = rowBase + v + 8 * half;
    if (row < N_LOC) {
      const float val = c[v] + bn;
      Y[(size_t)row * HHALF + n] = val > 0.f ? val : 0.f;
    }
  }
}

// ---------------------------------------------------------------------------
// logits[row] = h1[row, :] . W_h2 + b_h2   (32-MAC dot per row, L2-resident)
// ---------------------------------------------------------------------------
__global__ void head2_dot(const float* __restrict__ h1,
                          const float* __restrict__ W_h2,
                          const float* __restrict__ b_h2,
                          float* __restrict__ out) {
  const int row = blockIdx.x * blockDim.x + threadIdx.x;
  if (row >= N_LOC) return;
  float acc = b_h2[0];
#pragma unroll
  for (int j = 0; j < HHALF; ++j) acc += h1[(size_t)row * HHALF + j] * W_h2[j];
  out[row] = acc;
}

// ---------------------------------------------------------------------------
extern "C" void kernel_launch(void* const* d_in, const int* in_sizes, int n_in,
                              void* d_out, int out_size, void* d_ws, size_t ws_size,
                              hipStream_t stream) {
  const float*     loc_x = (const float*)d_in[0];
  const float*     evt_x = (const float*)d_in[1];
  const long long* ei    = (const long long*)d_in[2];
  const float*     W_loc = (const float*)d_in[3];
  const float*     b_loc = (const float*)d_in[4];
  const float*     W_evt = (const float*)d_in[5];
  const float*     b_evt = (const float*)d_in[6];
  const float*     W_l   = (const float*)d_in[7];
  const float*     b_l   = (const float*)d_in[8];
  const float*     W_r   = (const float*)d_in[9];
  const float*     W_h1  = (const float*)d_in[10];
  const float*     b_h1  = (const float*)d_in[11];
  const float*     W_h2  = (const float*)d_in[12];
  const float*     b_h2  = (const float*)d_in[13];
  float* out = (float*)d_out;

  // Workspace carving (floats): evt_h | loc_h | msg(->loc_h2) | cnt
  float* evt_h = (float*)d_ws;
  float* loc_h = evt_h + (size_t)N_EVT * HDIM;
  float* msg   = loc_h + (size_t)N_LOC * HDIM;
  float* cnt   = msg   + (size_t)N_LOC * HDIM;
  float* h1    = loc_h;  // reuse once loc_h is consumed by sage_combine

  // 1) zero accumulators (msg + cnt contiguous)
  const size_t nz = (size_t)N_LOC * HDIM + N_LOC;
  zero_kernel<<<(unsigned)((nz + 255) / 256), 256, 0, stream>>>(msg, nz);

  // 2) node projections (WMMA f32)
  proj_gemm_relu<<<N_EVT / 32, 256, 0, stream>>>(evt_x, W_evt, b_evt, evt_h, F_IN);
  proj_gemm_relu<<<N_LOC / 32, 256, 0, stream>>>(loc_x, W_loc, b_loc, loc_h, F_IN);

  // 3) edge scatter-mean numerators/counts
  const long long scatter_threads = (long long)NEDGE * 64;
  scatter_edges<<<(unsigned)((scatter_threads + 255) / 256), 256, 0, stream>>>(
      ei, evt_h, msg, cnt);

  // 4) SAGE combine (in place over msg -> loc_h2)
  sage_combine<<<N_LOC / 32, 256, 0, stream>>>(msg, cnt, loc_h, W_l, b_l, W_r, msg);

  // 5) head MLP
  head1_gemm_relu<<<(N_LOC + 63) / 64, 256, 0, stream>>>(msg, W_h1, b_h1, h1);
  head2_dot<<<(N_LOC + 255) / 256, 256, 0, stream>>>(h1, W_h2, b_h2, out);
}